// NSAB_5420248727612
// MI455X (gfx1250) — compile-verified
//
#include <hip/hip_runtime.h>

typedef __attribute__((ext_vector_type(2))) float v2f;
typedef __attribute__((ext_vector_type(8))) float v8f;
typedef __attribute__((ext_vector_type(4))) int   v4i;

#define TPB   256
#define VEC   4
#define NQ    12
#define NQP   16

#if __has_builtin(__builtin_amdgcn_global_load_async_to_lds_b128)
#define HAVE_ASYNC 1
#else
#define HAVE_ASYNC 0
#endif

#if __has_builtin(__builtin_amdgcn_s_wait_asynccnt)
#define WAIT_ASYNC(n) __builtin_amdgcn_s_wait_asynccnt(n)
#else
#define WAIT_ASYNC(n) asm volatile("s_wait_asynccnt %0" ::"i"(n) : "memory")
#endif

typedef __attribute__((address_space(1))) v4i gv4i;   // global b128
typedef __attribute__((address_space(3))) v4i lv4i;   // LDS b128

__global__ __launch_bounds__(TPB) void newton_sigmoid_rows(
    const float* __restrict__ x, const float* __restrict__ a,
    const float* __restrict__ b, const float* __restrict__ mean,
    const float* __restrict__ var, float* __restrict__ out,
    int Brows, int N)
{
    __shared__ float P[TPB][NQP];          // 16 KB: thread-partials (quant fastest)
#if HAVE_ASYNC
    __shared__ float4 stage[2][TPB];       // 8 KB: double-buffered async staging
#endif
    const int t   = threadIdx.x;
    const int row = blockIdx.x;
    const float ar = a[row];
    const float br = b[row];
    const float* xr = x + (size_t)row * (size_t)N;

    float q[NQ];
#pragma unroll
    for (int i = 0; i < NQ; ++i) q[i] = 0.0f;

    auto process = [&](float xv) {
        float u   = fmaf(ar, xv, br);
        float e   = __expf(-u);
        float s   = 1.0f / (1.0f + e);        // sigmoid
        float sp  = s * (1.0f - s);           // s'
        float spp = sp * (1.0f - 2.0f * s);   // s''
        float ssp = s * sp;                   // f * f'
        float g   = fmaf(s, spp, sp * sp);    // f'^2 + f*f''
        float x2  = xv * xv;
        q[0] += s;
        q[1]  = fmaf(sp,  xv, q[1]);
        q[2] += sp;
        q[3]  = fmaf(spp, x2, q[3]);
        q[4]  = fmaf(spp, xv, q[4]);
        q[5] += spp;
        q[6]  = fmaf(s,   s,  q[6]);
        q[7]  = fmaf(ssp, xv, q[7]);
        q[8] += ssp;
        q[9]  = fmaf(g,   x2, q[9]);
        q[10] = fmaf(g,   xv, q[10]);
        q[11] += g;
    };

    const int nchunks = N / (TPB * VEC);

#if HAVE_ASYNC
    auto issue = [&](int chunk, int buf) {
        const float* gp = xr + (size_t)(chunk * TPB + t) * VEC;
        __builtin_amdgcn_global_load_async_to_lds_b128(
            (gv4i*)gp, (lv4i*)&stage[buf][t], 0, 0);
    };
    if (nchunks > 0) issue(0, 0);
    for (int c = 0; c < nchunks; ++c) {
        if (c + 1 < nchunks) {
            issue(c + 1, (c + 1) & 1);
            WAIT_ASYNC(1);                 // current stage complete (in-order)
        } else {
            WAIT_ASYNC(0);
        }
        float4 v = stage[c & 1][t];
        process(v.x); process(v.y); process(v.z); process(v.w);
    }
#else
    for (int c = 0; c < nchunks; ++c) {
        float4 v = *(const float4*)(xr + (size_t)(c * TPB + t) * VEC);
        process(v.x); process(v.y); process(v.z); process(v.w);
    }
#endif
    // generic tail (N not multiple of 1024)
    for (int e = nchunks * TPB * VEC + t; e < N; e += TPB) process(xr[e]);

#pragma unroll
    for (int i = 0; i < NQ; ++i) P[t][i] = q[i];
#pragma unroll
    for (int i = NQ; i < NQP; ++i) P[t][i] = 0.0f;
    __syncthreads();

    // Wave 0: reduce 256 partials x 12 quantities with V_WMMA_F32_16X16X4_F32.
    // D(16x16) += A(16x4) x ones(4x16): A[m][k] = P[k0+k][m] (m = quantity).
    if (t < 32) {
        v8f c;
#pragma unroll
        for (int i = 0; i < 8; ++i) c[i] = 0.0f;
        const int qidx = t & 15;
        const int koff = (t >> 4) << 1;    // lanes 0-15 -> K{0,1}; 16-31 -> K{2,3}
        v2f bones; bones.x = 1.0f; bones.y = 1.0f;
#pragma unroll 4
        for (int k0 = 0; k0 < TPB; k0 += 4) {
            v2f av;
            av.x = P[k0 + koff + 0][qidx];
            av.y = P[k0 + koff + 1][qidx];
            c = __builtin_amdgcn_wmma_f32_16x16x4_f32(
                    false, av, false, bones, (short)0, c, false, false);
        }
        // Column N=0 sums: lane 0 holds M=0..7 in c[0..7]; lane 16 holds M=8..15.
        float sums[NQ];
#pragma unroll
        for (int i = 0; i < 8; ++i)  sums[i] = __shfl(c[i], 0, 32);
#pragma unroll
        for (int i = 8; i < NQ; ++i) sums[i] = __shfl(c[i - 8], 16, 32);

        if (t == 0) {
            const float invN = 1.0f / (float)N;
            float fm       = sums[0]  * invN;
            float dem_da   = sums[1]  * invN;
            float dem_db   = sums[2]  * invN;
            float d2em_da2 = sums[3]  * invN;
            float d2em_dab = sums[4]  * invN;
            float d2em_db2 = sums[5]  * invN;
            float mff      = sums[6]  * invN;
            float mfda     = sums[7]  * invN;
            float mfdb     = sums[8]  * invN;
            float mga2     = sums[9]  * invN;
            float mgab     = sums[10] * invN;
            float mgb2     = sums[11] * invN;

            float em = fm - mean[row];
            float ev = mff - fm * fm - var[row];
            float dev_da   = 2.0f * (mfda - fm * dem_da);
            float dev_db   = 2.0f * (mfdb - fm * dem_db);
            float d2ev_da2 = 2.0f * (mga2 - dem_da * dem_da - fm * d2em_da2);
            float d2ev_dab = 2.0f * (mgab - dem_da * dem_db - fm * d2em_dab);
            float d2ev_db2 = 2.0f * (mgb2 - dem_db * dem_db - fm * d2em_db2);

            float dl_da = 2.0f * (em * dem_da + ev * dev_da);
            float dl_db = 2.0f * (em * dem_db + ev * dev_db);
            float d2l_da2 = 2.0f * (dem_da * dem_da + em * d2em_da2 +
                                    dev_da * dev_da + ev * d2ev_da2);
            float d2l_dab = 2.0f * (dem_da * dem_db + em * d2em_dab +
                                    dev_da * dev_db + ev * d2ev_dab);
            float d2l_db2 = 2.0f * (dem_db * dem_db + em * d2em_db2 +
                                    dev_db * dev_db + ev * d2ev_db2);

            float den = d2l_da2 * d2l_db2 - d2l_dab * d2l_dab;
            float na  = (dl_da * d2l_db2 - dl_db * d2l_dab) / den;
            float nb  = (dl_db * d2l_da2 - dl_da * d2l_dab) / den;
            out[row]         = na;
            out[Brows + row] = nb;
        }
    }
}

extern "C" void kernel_launch(void* const* d_in, const int* in_sizes, int n_in,
                              void* d_out, int out_size, void* d_ws, size_t ws_size,
                              hipStream_t stream) {
    const float* x    = (const float*)d_in[0];
    const float* a    = (const float*)d_in[1];
    const float* b    = (const float*)d_in[2];
    const float* mean = (const float*)d_in[3];
    const float* var  = (const float*)d_in[4];
    float* out = (float*)d_out;
    const int B = in_sizes[1];              // (B,1) -> B elements
    const int N = in_sizes[0] / B;          // row length
    (void)n_in; (void)out_size; (void)d_ws; (void)ws_size;
    newton_sigmoid_rows<<<dim3(B), dim3(TPB), 0, stream>>>(x, a, b, mean, var, out, B, N);
}